// GINE_43654047596959
// MI455X (gfx1250) — compile-verified
//
#include <hip/hip_runtime.h>
#include <hip/hip_bf16.h>

#define N_NODES 50000
#define N_EDGES 1600000
#define HID 256
#define EDGE_F 7

typedef __attribute__((ext_vector_type(2))) float v2f;
typedef __attribute__((ext_vector_type(8))) float v8f;

// ---------------- utility kernels ----------------

__global__ __launch_bounds__(256) void zero_f32(float* __restrict__ p, long n) {
    long i = (long)blockIdx.x * blockDim.x + threadIdx.x;
    if (i < n) p[i] = 0.0f;
}

__global__ __launch_bounds__(256) void relu_copy(const float* __restrict__ in,
                                                 float* __restrict__ out, long n) {
    long i = (long)blockIdx.x * blockDim.x + threadIdx.x;
    if (i < n) out[i] = fmaxf(in[i], 0.0f);
}

// ---------------- layer 1 (NODE_F = 2) ----------------

// per-edge: emb = edge_attr @ We1 + be1 (7->2); msg = relu(x[src] + emb); atomic scatter
__global__ __launch_bounds__(256) void msg_scatter_l1(
    const float* __restrict__ x, const long long* __restrict__ ei,
    const float* __restrict__ ea, const float* __restrict__ We1,
    const float* __restrict__ be1, float* __restrict__ aggr1, int E) {
    int e = blockIdx.x * blockDim.x + threadIdx.x;
    if (e >= E) return;
    float emb0 = be1[0], emb1 = be1[1];
#pragma unroll
    for (int k = 0; k < EDGE_F; ++k) {
        float a = ea[(long)e * EDGE_F + k];
        emb0 = fmaf(a, We1[k * 2 + 0], emb0);
        emb1 = fmaf(a, We1[k * 2 + 1], emb1);
    }
    long s = (long)ei[e];
    long d = (long)ei[(long)N_EDGES + e];
    float m0 = fmaxf(x[s * 2 + 0] + emb0, 0.0f);
    float m1 = fmaxf(x[s * 2 + 1] + emb1, 0.0f);
    atomicAdd(&aggr1[d * 2 + 0], m0);
    atomicAdd(&aggr1[d * 2 + 1], m1);
}

// h[n,c] = b1[c] + (x[n,0]+aggr1[n,0])*W1[0,c] + (x[n,1]+aggr1[n,1])*W1[1,c]
__global__ __launch_bounds__(256) void dense_l1(
    const float* __restrict__ x, const float* __restrict__ aggr1,
    const float* __restrict__ W1, const float* __restrict__ b1,
    float* __restrict__ h, int N) {
    long idx = (long)blockIdx.x * blockDim.x + threadIdx.x;
    if (idx >= (long)N * HID) return;
    long n = idx >> 8;
    int c = (int)(idx & 255);
    float s0 = x[n * 2 + 0] + aggr1[n * 2 + 0];
    float s1 = x[n * 2 + 1] + aggr1[n * 2 + 1];
    h[idx] = fmaf(s0, W1[c], fmaf(s1, W1[HID + c], b1[c]));
}

// ---------------- layers 2..3: fused edge embedding + message + scatter ----------------

// thread c handles channel c for EPB edges; We2 [7,256] staged in LDS.
__global__ __launch_bounds__(256) void msg_scatter_l2(
    const float* __restrict__ hr, const long long* __restrict__ ei,
    const float* __restrict__ ea, const float* __restrict__ We2,
    const float* __restrict__ be2, float* __restrict__ aggr, int E, int EPB) {
    __shared__ float Ws[EDGE_F * HID];
    __shared__ float bs[HID];
    int c = threadIdx.x;
#pragma unroll
    for (int k = 0; k < EDGE_F; ++k) Ws[k * HID + c] = We2[k * HID + c];
    bs[c] = be2[c];
    __syncthreads();

    long e0 = (long)blockIdx.x * EPB;
    long e1 = e0 + EPB;
    if (e1 > E) e1 = E;
    float bias = bs[c];
    for (long e = e0; e < e1; ++e) {
        // prefetch edge_attr a few edges ahead (global_prefetch_b8)
        if (e + 8 < e1) __builtin_prefetch(&ea[(e + 8) * EDGE_F], 0, 0);
        float emb = bias;
#pragma unroll
        for (int k = 0; k < EDGE_F; ++k)
            emb = fmaf(ea[e * EDGE_F + k], Ws[k * HID + c], emb);
        long s = (long)ei[e];
        long d = (long)ei[(long)N_EDGES + e];
        float msg = fmaxf(hr[s * HID + c] + emb, 0.0f);
        atomicAdd(&aggr[d * HID + c], msg);
    }
}

// ---------------- WMMA f32 GEMM: out = (hr + aggr) @ W + b ----------------
// One wave32 computes a 16-row x 64-col strip (4 col tiles), K=256 via
// V_WMMA_F32_16X16X4_F32. A fragment (16x4) reused across the 4 col tiles.
__global__ __launch_bounds__(256) void gemm_wmma_f32(
    const float* __restrict__ hr, const float* __restrict__ aggr,
    const float* __restrict__ W, const float* __restrict__ bvec,
    float* __restrict__ out, int N) {
    int wave = (blockIdx.x * blockDim.x + threadIdx.x) >> 5;
    int lane = threadIdx.x & 31;
    int rowTiles = N >> 4;              // 3125
    int colGroup = wave & 3;            // 4 groups x 4 tiles = 16 col tiles
    int rowTile = wave >> 2;
    if (rowTile >= rowTiles) return;    // wave-uniform exit; EXEC stays all-ones

    int m0 = rowTile << 4;
    int n0base = colGroup << 6;
    int m = lane & 15;                  // A: row within tile
    int kk = (lane >> 4) << 1;          // A/B: K sub-offset {0,2}
    int n = lane & 15;                  // B/C/D: col within tile

    v8f acc0 = {}, acc1 = {}, acc2 = {}, acc3 = {};
    const long arow = (long)(m0 + m) * HID;
    for (int k0 = 0; k0 < HID; k0 += 4) {
        int k = k0 + kk;
        v2f a;
        a.x = hr[arow + k]     + aggr[arow + k];
        a.y = hr[arow + k + 1] + aggr[arow + k + 1];
        const float* Wk0 = W + (long)k * HID;
        const float* Wk1 = W + (long)(k + 1) * HID;
        v2f b0; b0.x = Wk0[n0base + n];      b0.y = Wk1[n0base + n];
        v2f b1; b1.x = Wk0[n0base + 16 + n]; b1.y = Wk1[n0base + 16 + n];
        v2f b2; b2.x = Wk0[n0base + 32 + n]; b2.y = Wk1[n0base + 32 + n];
        v2f b3; b3.x = Wk0[n0base + 48 + n]; b3.y = Wk1[n0base + 48 + n];
        acc0 = __builtin_amdgcn_wmma_f32_16x16x4_f32(false, a, false, b0, (short)0, acc0, false, false);
        acc1 = __builtin_amdgcn_wmma_f32_16x16x4_f32(false, a, false, b1, (short)0, acc1, false, false);
        acc2 = __builtin_amdgcn_wmma_f32_16x16x4_f32(false, a, false, b2, (short)0, acc2, false, false);
        acc3 = __builtin_amdgcn_wmma_f32_16x16x4_f32(false, a, false, b3, (short)0, acc3, false, false);
    }

    // D layout: VGPR v holds row m0 + v + 8*(lane>>4), col n0 + (lane&15)
    int mrow0 = m0 + ((lane >> 4) << 3);
#pragma unroll
    for (int t = 0; t < 4; ++t) {
        const v8f& acc = (t == 0) ? acc0 : (t == 1) ? acc1 : (t == 2) ? acc2 : acc3;
        int nn = n0base + (t << 4) + n;
        float bias = bvec[nn];
#pragma unroll
        for (int v = 0; v < 8; ++v)
            out[(long)(mrow0 + v) * HID + nn] = acc[v] + bias;
    }
}

// ---------------- final head: sigmoid(relu(h) @ Wend + bend) ----------------

__global__ __launch_bounds__(256) void final_head(
    const float* __restrict__ hr, const float* __restrict__ Wend,
    const float* __restrict__ bend, float* __restrict__ out, int N) {
    int nIdx = blockIdx.x * blockDim.x + threadIdx.x;
    if (nIdx >= N) return;
    float s = bend[0];
    const float* row = hr + (long)nIdx * HID;
#pragma unroll 8
    for (int c = 0; c < HID; ++c) s = fmaf(row[c], Wend[c], s);
    out[nIdx] = 1.0f / (1.0f + expf(-s));
}

// ---------------- launch ----------------

extern "C" void kernel_launch(void* const* d_in, const int* in_sizes, int n_in,
                              void* d_out, int out_size, void* d_ws, size_t ws_size,
                              hipStream_t stream) {
    const float*     x    = (const float*)d_in[0];
    const long long* ei   = (const long long*)d_in[1];   // int64 [2, E]
    const float*     ea   = (const float*)d_in[2];       // [E, 7]
    const float*     We1  = (const float*)d_in[3];
    const float*     be1  = (const float*)d_in[4];
    const float*     W1   = (const float*)d_in[5];
    const float*     b1   = (const float*)d_in[6];
    const float*     We2  = (const float*)d_in[7];
    const float*     be2  = (const float*)d_in[8];
    const float*     W2   = (const float*)d_in[9];
    const float*     b2   = (const float*)d_in[10];
    const float*     Wend = (const float*)d_in[11];
    const float*     bend = (const float*)d_in[12];
    float* outp = (float*)d_out;

    const long NH = (long)N_NODES * HID;            // 12.8M
    float* h    = (float*)d_ws;                     // [N, 256]
    float* hr   = h + NH;                           // [N, 256]
    float* aggr = hr + NH;                          // [N, 256] (layer1 uses first N*2)

    const int elemBlocks = (int)((NH + 255) / 256); // 50000
    const int EPB = 64;
    const int msgBlocks = (N_EDGES + EPB - 1) / EPB;            // 25000
    const int gemmBlocks = ((N_NODES / 16) * 4 + 7) / 8;        // 12500 waves / 8

    // ---- layer 1 ----
    zero_f32<<<(N_NODES * 2 + 255) / 256, 256, 0, stream>>>(aggr, (long)N_NODES * 2);
    msg_scatter_l1<<<(N_EDGES + 255) / 256, 256, 0, stream>>>(x, ei, ea, We1, be1, aggr, N_EDGES);
    dense_l1<<<elemBlocks, 256, 0, stream>>>(x, aggr, W1, b1, h, N_NODES);

    // ---- layers 2..3 ----
    for (int l = 0; l < 2; ++l) {
        relu_copy<<<elemBlocks, 256, 0, stream>>>(h, hr, NH);
        zero_f32<<<elemBlocks, 256, 0, stream>>>(aggr, NH);
        msg_scatter_l2<<<msgBlocks, 256, 0, stream>>>(hr, ei, ea, We2, be2, aggr, N_EDGES, EPB);
        gemm_wmma_f32<<<gemmBlocks, 256, 0, stream>>>(hr, aggr, W2, b2, h, N_NODES);
    }

    // ---- head ----
    relu_copy<<<elemBlocks, 256, 0, stream>>>(h, hr, NH);
    final_head<<<(N_NODES + 255) / 256, 256, 0, stream>>>(hr, Wend, bend, outp, N_NODES);
}